// WaveletSkipConnection_48335561949944
// MI455X (gfx1250) — compile-verified
//
#include <hip/hip_runtime.h>
#include <cstddef>
#include <cstdint>

typedef float v2f __attribute__((ext_vector_type(2)));
typedef float v8f __attribute__((ext_vector_type(8)));

#define B_   8
#define C_   256
#define H_   128
#define W_   128
#define HC   64
#define WC   64
#define CNT_ ((float)(B_ * H_ * W_))

// workspace layout (in floats)
#define WT_OFF  ((size_t)0)                       // transposed weights: [c_full(1024)][o(256)]
#define WT_SZ   ((size_t)1024 * 256)
#define D_OFF   (WT_OFF + WT_SZ)                  // coarse detail GEMM out: [b][kh][o][kw]
#define D_SZ    ((size_t)B_ * HC * C_ * WC)
#define SUM_OFF (D_OFF + D_SZ)                    // per-channel sum  [256]
#define SQ_OFF  (SUM_OFF + 256)                   // per-channel sum2 [256]
#define SB_OFF  (SQ_OFF + 256)                    // scale/bias pairs [512]

__device__ __forceinline__ v8f wmma_f32_k4(v2f a, v2f b, v8f c) {
    // D = A(16x4) * B(4x16) + C(16x16), full f32
    return __builtin_amdgcn_wmma_f32_16x16x4_f32(false, a, false, b, (short)0, c,
                                                 false, false);
}

// ---------------------------------------------------------------------------
// Kernel 0: transpose conv_w (256x1024, row-major [o][c]) -> wt [c][o];
//           zero the global stat accumulators.
// ---------------------------------------------------------------------------
__global__ void __launch_bounds__(256) prep_kernel(const float* __restrict__ cw,
                                                   float* __restrict__ ws) {
    const int cfull = blockIdx.x;    // 0..1023
    const int o     = threadIdx.x;   // 0..255
    ws[WT_OFF + (size_t)cfull * 256 + o] = cw[(size_t)o * 1024 + cfull];
    if (blockIdx.x == 0) {
        ws[SUM_OFF + o] = 0.0f;
        ws[SQ_OFF  + o] = 0.0f;
    }
}

// ---------------------------------------------------------------------------
// Kernel 1: coarse detail GEMM.
// For one (b, kh): d[b][kh][o][kw] = sum_c WH[o][c]*cH[c][kw] + WV*cV + WD*cD
// Haar details computed on the fly into LDS; consumed as WMMA B-fragments.
// M = o (16 tiles of 16), N = kw (4 tiles of 16), K = 768 in 4-wide steps.
// ---------------------------------------------------------------------------
__global__ void __launch_bounds__(256) coarse_kernel(const float* __restrict__ x,
                                                     float* __restrict__ ws) {
    const int b    = blockIdx.x / HC;
    const int kh   = blockIdx.x % HC;
    const int tid  = threadIdx.x;
    const int lane = tid & 31;
    const int wave = tid >> 5;       // 0..7
    const int lm   = lane & 15;
    const int half = lane >> 4;

    __shared__ float sH[8][64];
    __shared__ float sV[8][64];
    __shared__ float sD[8][64];

    const float* wt   = ws + WT_OFF;
    float*       dOut = ws + D_OFF;

    v8f acc[2][4];
#pragma unroll
    for (int i = 0; i < 2; ++i)
#pragma unroll
        for (int j = 0; j < 4; ++j)
#pragma unroll
            for (int e = 0; e < 8; ++e) acc[i][j][e] = 0.0f;

    const int ci = tid >> 5;  // channel-in-chunk this thread stages (0..7)
    const int wp = tid & 31;  // coarse-column pair (covers kw = 2wp, 2wp+1)

    for (int c0 = 0; c0 < 256; c0 += 8) {
        __syncthreads();  // previous iteration's LDS reads complete
        // ---- stage: Haar details for channels c0..c0+7 of row pair 2kh,2kh+1
        const float* xr = x + (((size_t)(b * C_ + c0 + ci) * H_ + 2 * kh) * W_);
        float4 r0 = *(const float4*)(xr + 4 * wp);        // p0 q0 p1 q1
        float4 r1 = *(const float4*)(xr + W_ + 4 * wp);   // r0 s0 r1 s1
        sH[ci][2 * wp]     = 0.5f * (r0.x + r0.y - r1.x - r1.y);
        sV[ci][2 * wp]     = 0.5f * (r0.x - r0.y + r1.x - r1.y);
        sD[ci][2 * wp]     = 0.5f * (r0.x - r0.y - r1.x + r1.y);
        sH[ci][2 * wp + 1] = 0.5f * (r0.z + r0.w - r1.z - r1.w);
        sV[ci][2 * wp + 1] = 0.5f * (r0.z - r0.w + r1.z - r1.w);
        sD[ci][2 * wp + 1] = 0.5f * (r0.z - r0.w - r1.z + r1.w);
        __syncthreads();

        // ---- GEMM over the 8-channel chunk: 2 K-groups x 3 detail planes
#pragma unroll
        for (int cg = 0; cg < 8; cg += 4) {
#pragma unroll
            for (int det = 0; det < 3; ++det) {
                const float(*sx)[64] = (det == 0) ? sH : (det == 1) ? sV : sD;
                const int cfull = 256 * (1 + det) + c0 + cg + 2 * half;  // + j
                v2f a0, a1;  // A fragments for the wave's two o-tiles
                a0[0] = wt[(size_t)cfull * 256 + wave * 32 + lm];
                a0[1] = wt[(size_t)(cfull + 1) * 256 + wave * 32 + lm];
                a1[0] = wt[(size_t)cfull * 256 + wave * 32 + 16 + lm];
                a1[1] = wt[(size_t)(cfull + 1) * 256 + wave * 32 + 16 + lm];
#pragma unroll
                for (int nt = 0; nt < 4; ++nt) {
                    v2f bf;
                    bf[0] = sx[cg + 2 * half][nt * 16 + lm];
                    bf[1] = sx[cg + 2 * half + 1][nt * 16 + lm];
                    acc[0][nt] = wmma_f32_k4(a0, bf, acc[0][nt]);
                    acc[1][nt] = wmma_f32_k4(a1, bf, acc[1][nt]);
                }
            }
        }
    }

    // ---- epilogue: d[b][kh][o][kw], lanes write 16 consecutive kw (64B)
#pragma unroll
    for (int mt = 0; mt < 2; ++mt)
#pragma unroll
        for (int nt = 0; nt < 4; ++nt)
#pragma unroll
            for (int r = 0; r < 8; ++r) {
                const int o  = wave * 32 + mt * 16 + r + 8 * half;
                const int kw = nt * 16 + lm;
                dOut[(((size_t)(b * HC + kh)) * C_ + o) * WC + kw] = acc[mt][nt][r];
            }
}

// ---------------------------------------------------------------------------
// Kernel 2: main GEMM  y = Wx * x  + bilinear_up(d), plus BN partial stats.
// One block: (b, h, 64-pixel w segment) x all 256 output channels.
// x tile (256 x 64 f32 = 64KB) is staged into LDS once with
// GLOBAL_LOAD_ASYNC_TO_LDS_B128 (ASYNCcnt path), then the K-loop reads
// B fragments with ds_load_2addr and A fragments from L1/L2-resident wt.
// ---------------------------------------------------------------------------
__global__ void __launch_bounds__(256) main_kernel(const float* __restrict__ x,
                                                   float* __restrict__ y,
                                                   float* __restrict__ ws) {
    const int wseg = blockIdx.x & 1;
    const int h    = (blockIdx.x >> 1) & (H_ - 1);
    const int b    = blockIdx.x >> 8;
    const int w0   = wseg * 64;

    const int tid  = threadIdx.x;
    const int lane = tid & 31;
    const int wave = tid >> 5;
    const int lm   = lane & 15;
    const int half = lane >> 4;

    const float* wt = ws + WT_OFF;
    const float* d  = ws + D_OFF;

    __shared__ float sx[C_][64];     // x tile: [c][w-w0], 64KB
    __shared__ float redS[256];
    __shared__ float redQ[256];

    v8f acc[2][4];
#pragma unroll
    for (int i = 0; i < 2; ++i)
#pragma unroll
        for (int j = 0; j < 4; ++j)
#pragma unroll
            for (int e = 0; e < 8; ++e) acc[i][j][e] = 0.0f;

    // ---- async stage: 4096 x 16B transfers, 16 per thread, coalesced.
    {
        const float*   gbase = x + (size_t)b * C_ * H_ * W_ + (size_t)h * W_ + w0;
        const unsigned lbase = (unsigned)(uintptr_t)(&sx[0][0]);
#pragma unroll
        for (int it = 0; it < 16; ++it) {
            const int      i   = it * 256 + tid;   // 16B-transfer index
            const int      c   = i >> 4;           // channel
            const int      seg = i & 15;           // 16B segment within row
            const float*   gp  = gbase + (size_t)c * (H_ * W_) + seg * 4;
            const unsigned lp  = lbase + (unsigned)i * 16;
            asm volatile("global_load_async_to_lds_b128 %0, %1, off"
                         :: "v"(lp), "v"(gp)
                         : "memory");
        }
        asm volatile("s_wait_asynccnt 0x0" ::: "memory");
        __syncthreads();
    }

    // ---- K loop: A from global wt (4 b32), B from LDS (4 ds_load_2addr)
    for (int k0 = 0; k0 < 256; k0 += 4) {
        const int c = k0 + 2 * half;
        v2f a0, a1;  // A = weights, M = o
        a0[0] = wt[(size_t)c * 256 + wave * 32 + lm];
        a0[1] = wt[(size_t)(c + 1) * 256 + wave * 32 + lm];
        a1[0] = wt[(size_t)c * 256 + wave * 32 + 16 + lm];
        a1[1] = wt[(size_t)(c + 1) * 256 + wave * 32 + 16 + lm];
        const float* row = &sx[c][0];
#pragma unroll
        for (int nt = 0; nt < 4; ++nt) {
            v2f bf;  // B = x pixels, N = w   (rows c and c+1 are 64 floats apart)
            bf[0] = row[nt * 16 + lm];
            bf[1] = row[64 + nt * 16 + lm];
            acc[0][nt] = wmma_f32_k4(a0, bf, acc[0][nt]);
            acc[1][nt] = wmma_f32_k4(a1, bf, acc[1][nt]);
        }
    }

    // bilinear 2x upsample taps along h (uniform for the block)
    const int hk = h >> 1;
    int   kh0, kh1;
    float wh0, wh1;
    if (h & 1) { kh0 = hk;                   wh0 = 0.75f; kh1 = (hk + 1 < HC) ? hk + 1 : HC - 1; wh1 = 0.25f; }
    else       { kh0 = (hk > 0) ? hk - 1 : 0; wh0 = 0.25f; kh1 = hk;                              wh1 = 0.75f; }

#pragma unroll
    for (int mt = 0; mt < 2; ++mt) {
#pragma unroll
        for (int r = 0; r < 8; ++r) {
            const int o = wave * 32 + mt * 16 + r + 8 * half;
            const float* d0 = d + (((size_t)(b * HC + kh0)) * C_ + o) * WC;
            const float* d1 = d + (((size_t)(b * HC + kh1)) * C_ + o) * WC;
            float s = 0.0f, q = 0.0f;
#pragma unroll
            for (int nt = 0; nt < 4; ++nt) {
                const int w  = w0 + nt * 16 + lm;
                const int wk = w >> 1;
                int   kw0, kw1;
                float a0w, a1w;
                if (w & 1) { kw0 = wk;                   a0w = 0.75f; kw1 = (wk + 1 < WC) ? wk + 1 : WC - 1; a1w = 0.25f; }
                else       { kw0 = (wk > 0) ? wk - 1 : 0; a0w = 0.25f; kw1 = wk;                              a1w = 0.75f; }
                const float up = wh0 * (a0w * d0[kw0] + a1w * d0[kw1]) +
                                 wh1 * (a0w * d1[kw0] + a1w * d1[kw1]);
                const float yv = acc[mt][nt][r] + up;
                y[(((size_t)(b * C_ + o)) * H_ + h) * W_ + w] = yv;
                s += yv;
                q += yv * yv;
            }
            // reduce the 16-lane (lm) group; unique writer per o in this block
#pragma unroll
            for (int m = 1; m < 16; m <<= 1) {
                s += __shfl_xor(s, m, 32);
                q += __shfl_xor(q, m, 32);
            }
            if (lm == 0) { redS[o] = s; redQ[o] = q; }
        }
    }
    __syncthreads();
    atomicAdd(&ws[SUM_OFF + tid], redS[tid]);
    atomicAdd(&ws[SQ_OFF + tid], redQ[tid]);
}

// ---------------------------------------------------------------------------
// Kernel 3: finalize BN statistics -> per-channel scale/bias
// ---------------------------------------------------------------------------
__global__ void stats_kernel(const float* __restrict__ gamma,
                             const float* __restrict__ beta,
                             float* __restrict__ ws) {
    const int o = threadIdx.x;
    const float mean = ws[SUM_OFF + o] / CNT_;
    const float var  = ws[SQ_OFF + o] / CNT_ - mean * mean;
    const float sc   = gamma[o] * rsqrtf(var + 1e-5f);
    ws[SB_OFF + 2 * o]     = sc;
    ws[SB_OFF + 2 * o + 1] = beta[o] - mean * sc;
}

// ---------------------------------------------------------------------------
// Kernel 4: in-place normalize + ReLU on y (float4 along w)
// ---------------------------------------------------------------------------
__global__ void __launch_bounds__(256) norm_kernel(float* __restrict__ y,
                                                   const float* __restrict__ ws) {
    const size_t i4 = (size_t)blockIdx.x * 256 + threadIdx.x;
    const size_t e  = i4 * 4;
    const int    o  = (int)((e >> 14) & 255);  // H*W = 16384
    const float  sc = ws[SB_OFF + 2 * o];
    const float  bi = ws[SB_OFF + 2 * o + 1];
    float4 v = *(float4*)(y + e);
    v.x = fmaxf(v.x * sc + bi, 0.0f);
    v.y = fmaxf(v.y * sc + bi, 0.0f);
    v.z = fmaxf(v.z * sc + bi, 0.0f);
    v.w = fmaxf(v.w * sc + bi, 0.0f);
    *(float4*)(y + e) = v;
}

extern "C" void kernel_launch(void* const* d_in, const int* in_sizes, int n_in,
                              void* d_out, int out_size, void* d_ws, size_t ws_size,
                              hipStream_t stream) {
    const float* x     = (const float*)d_in[0];
    const float* cw    = (const float*)d_in[1];
    const float* gamma = (const float*)d_in[2];
    const float* beta  = (const float*)d_in[3];
    float*       y     = (float*)d_out;
    float*       ws    = (float*)d_ws;

    prep_kernel<<<1024, 256, 0, stream>>>(cw, ws);
    coarse_kernel<<<B_ * HC, 256, 0, stream>>>(x, ws);
    main_kernel<<<B_ * H_ * (W_ / 64), 256, 0, stream>>>(x, y, ws);
    stats_kernel<<<1, 256, 0, stream>>>(gamma, beta, ws);
    norm_kernel<<<(B_ * C_ * H_ * W_) / (4 * 256), 256, 0, stream>>>(y, ws);
}